// GCNNet_58385785422144
// MI455X (gfx1250) — compile-verified
//
#include <hip/hip_runtime.h>
#include <hip/hip_bf16.h>

#define NODES   50000
#define EDGES   800000
#define DIN     100
#define HIDN    128
#define NLAYERS 3
#define NGRAPHS 64

typedef __attribute__((ext_vector_type(2))) float v2f;
typedef __attribute__((ext_vector_type(8))) float v8f;

__device__ __forceinline__ void atomAddF(float* p, float v) {
    __hip_atomic_fetch_add(p, v, __ATOMIC_RELAXED, __HIP_MEMORY_SCOPE_AGENT);
}

// ---------------------------------------------------------------------------
// C[M x 128] = A[M x K] @ B[K x 128] (+ bias) using V_WMMA_F32_16X16X4_F32.
// grid.x = M/16 (row tile), 8 waves per block = 8 column tiles (N = 128).
// A-frag (16x4 f32): lane<16 -> row=lane, K={k,k+1}; lane>=16 -> K={k+2,k+3}.
// B-frag (4x16 f32): mirrored (half selects K offset 0 / 2).
// C/D: VGPR r -> row = r + 8*(lane>=16), col = lane&15.
// ---------------------------------------------------------------------------
template<int K, bool ADD_BIAS>
__global__ __launch_bounds__(256) void gemm_wmma_f32(
    const float* __restrict__ A, const float* __restrict__ B,
    const float* __restrict__ bias, float* __restrict__ C) {
    const int lane = threadIdx.x & 31;
    const int wave = threadIdx.x >> 5;
    const int m0   = blockIdx.x << 4;
    const int n0   = wave << 4;
    const int half = lane >> 4;     // 0: lanes 0-15, 1: lanes 16-31
    const int l15  = lane & 15;

    const float* arow = A + (size_t)(m0 + l15) * K;   // this lane's A row
    const float* bcol = B + n0 + l15;                 // this lane's B column

    v8f acc = {};
#pragma unroll 8
    for (int k = 0; k < K; k += 4) {
        const int ka = k + (half << 1);
        v2f a = *(const v2f*)(arow + ka);             // contiguous, 8B aligned
        v2f b;
        b.x = bcol[(size_t)ka * HIDN];
        b.y = bcol[(size_t)(ka + 1) * HIDN];
        acc = __builtin_amdgcn_wmma_f32_16x16x4_f32(
            false, a, false, b, (short)0, acc, false, false);
    }

    float badd = 0.0f;
    if constexpr (ADD_BIAS) badd = bias[n0 + l15];
    float* crow = C + (size_t)(m0 + (half << 3)) * HIDN + n0 + l15;
#pragma unroll
    for (int r = 0; r < 8; ++r)
        crow[(size_t)r * HIDN] = acc[r] + badd;
}

// --------------------------- degree / norm ---------------------------------
__global__ void k_deg_init(float* __restrict__ deg) {
    int i = blockIdx.x * blockDim.x + threadIdx.x;
    if (i < NODES) deg[i] = 1.0f;                      // self loop
}
__global__ void k_deg_edges(const int* __restrict__ dst, float* __restrict__ deg) {
    int e = blockIdx.x * blockDim.x + threadIdx.x;
    if (e < EDGES) atomAddF(&deg[dst[e]], 1.0f);
}
__global__ void k_dinv(float* __restrict__ d) {
    int i = blockIdx.x * blockDim.x + threadIdx.x;
    if (i < NODES) d[i] = rsqrtf(d[i]);                // deg >= 1 always
}

// ----------------------- aggregation (message passing) ---------------------
// out[i] = hw[i] * dinv[i]^2   (self-loop term; also zero-initializes out)
__global__ __launch_bounds__(256) void k_agg_self(
    const float* __restrict__ hw, const float* __restrict__ dinv,
    float* __restrict__ out) {
    int t = blockIdx.x * blockDim.x + threadIdx.x;     // NODES * 32 threads
    int node = t >> 5;
    int c = (t & 31) << 2;
    if (node >= NODES) return;
    float w = dinv[node]; w *= w;
    float4 v = *(const float4*)(hw + (size_t)node * HIDN + c);
    v.x *= w; v.y *= w; v.z *= w; v.w *= w;
    *(float4*)(out + (size_t)node * HIDN + c) = v;
}

// out[dst[e]] += hw[src[e]] * dinv[src]*dinv[dst]   — one wave per edge
__global__ __launch_bounds__(256) void k_agg_edges(
    const float* __restrict__ hw, const int* __restrict__ src,
    const int* __restrict__ dst, const float* __restrict__ dinv,
    float* __restrict__ out) {
    int e = (blockIdx.x * blockDim.x + threadIdx.x) >> 5;
    int lane = threadIdx.x & 31;
    if (e >= EDGES) return;
    int s = src[e], d = dst[e];
    float w = dinv[s] * dinv[d];
    float4 v = *(const float4*)(hw + (size_t)s * HIDN + (lane << 2));
    float* o = out + (size_t)d * HIDN + (lane << 2);
    atomAddF(o + 0, v.x * w);
    atomAddF(o + 1, v.y * w);
    atomAddF(o + 2, v.z * w);
    atomAddF(o + 3, v.w * w);
}

__global__ void k_bias_relu(float* __restrict__ h, const float* __restrict__ b) {
    int i = blockIdx.x * blockDim.x + threadIdx.x;     // NODES * HIDN
    if (i < NODES * HIDN) {
        float v = h[i] + b[i & (HIDN - 1)];
        h[i] = v > 0.0f ? v : 0.0f;
    }
}

// ------------------------------- pooling -----------------------------------
__global__ void k_pool_zero(float* __restrict__ pooled, float* __restrict__ cnt) {
    int i = blockIdx.x * blockDim.x + threadIdx.x;
    if (i < NGRAPHS * HIDN) pooled[i] = 0.0f;
    if (i < NGRAPHS) cnt[i] = 0.0f;
}
__global__ __launch_bounds__(256) void k_pool_scatter(
    const float* __restrict__ h, const int* __restrict__ batch,
    float* __restrict__ pooled, float* __restrict__ cnt) {
    int t = blockIdx.x * blockDim.x + threadIdx.x;     // NODES * 32 threads
    int node = t >> 5;
    int lane = t & 31;
    if (node >= NODES) return;
    int g = batch[node];
    float4 v = *(const float4*)(h + (size_t)node * HIDN + (lane << 2));
    float* p = pooled + (size_t)g * HIDN + (lane << 2);
    atomAddF(p + 0, v.x);
    atomAddF(p + 1, v.y);
    atomAddF(p + 2, v.z);
    atomAddF(p + 3, v.w);
    if (lane == 0) atomAddF(&cnt[g], 1.0f);
}
__global__ void k_pool_out(const float* __restrict__ pooled,
                           const float* __restrict__ cnt,
                           const float* __restrict__ Wreg,
                           const float* __restrict__ breg,
                           float* __restrict__ out) {
    int g = threadIdx.x;
    if (g >= NGRAPHS) return;
    float inv = 1.0f / fmaxf(cnt[g], 1.0f);
    float s = 0.0f;
    for (int c = 0; c < HIDN; ++c)
        s += pooled[(size_t)g * HIDN + c] * inv * Wreg[c];
    out[g] = s + breg[0];
}

// ---------------------------------------------------------------------------
extern "C" void kernel_launch(void* const* d_in, const int* in_sizes, int n_in,
                              void* d_out, int out_size, void* d_ws, size_t ws_size,
                              hipStream_t stream) {
    const float* x     = (const float*)d_in[0];
    const int*   edge  = (const int*)d_in[1];
    const int*   src   = edge;
    const int*   dst   = edge + EDGES;
    const int*   batch = (const int*)d_in[2];
    const float* W_enc = (const float*)d_in[3];
    const float* b_enc = (const float*)d_in[4];
    const float* gcn_W = (const float*)d_in[5];
    const float* gcn_b = (const float*)d_in[6];
    const float* W_reg = (const float*)d_in[7];
    const float* b_reg = (const float*)d_in[8];
    float* out = (float*)d_out;

    // workspace carve-up
    float* ws = (float*)d_ws;
    float* h      = ws;                                   // NODES*HIDN
    float* hw     = h  + (size_t)NODES * HIDN;            // NODES*HIDN
    float* dinv   = hw + (size_t)NODES * HIDN;            // NODES
    float* pooled = dinv + NODES;                         // NGRAPHS*HIDN
    float* cnt    = pooled + NGRAPHS * HIDN;              // NGRAPHS

    const int TPB = 256;
    const int rowTiles = NODES / 16;                      // 3125, exact

    // degree -> dinv
    k_deg_init <<<(NODES + TPB - 1) / TPB, TPB, 0, stream>>>(dinv);
    k_deg_edges<<<(EDGES + TPB - 1) / TPB, TPB, 0, stream>>>(dst, dinv);
    k_dinv     <<<(NODES + TPB - 1) / TPB, TPB, 0, stream>>>(dinv);

    // encoder: h = x @ W_enc + b_enc
    gemm_wmma_f32<DIN, true><<<rowTiles, TPB, 0, stream>>>(x, W_enc, b_enc, h);

    // GCN layers
    for (int l = 0; l < NLAYERS; ++l) {
        gemm_wmma_f32<HIDN, false><<<rowTiles, TPB, 0, stream>>>(
            h, gcn_W + (size_t)l * HIDN * HIDN, nullptr, hw);
        k_agg_self <<<(NODES * 32 + TPB - 1) / TPB, TPB, 0, stream>>>(hw, dinv, h);
        k_agg_edges<<<(EDGES * 32 + TPB - 1) / TPB, TPB, 0, stream>>>(hw, src, dst, dinv, h);
        k_bias_relu<<<(NODES * HIDN + TPB - 1) / TPB, TPB, 0, stream>>>(
            h, gcn_b + (size_t)l * HIDN);
    }

    // global mean pool + regression head
    k_pool_zero   <<<(NGRAPHS * HIDN + TPB - 1) / TPB, TPB, 0, stream>>>(pooled, cnt);
    k_pool_scatter<<<(NODES * 32 + TPB - 1) / TPB, TPB, 0, stream>>>(h, batch, pooled, cnt);
    k_pool_out    <<<1, NGRAPHS, 0, stream>>>(pooled, cnt, W_reg, b_reg, out);

    (void)in_sizes; (void)n_in; (void)out_size; (void)ws_size;
}